// UpFirDnUpsample_6073083756672
// MI455X (gfx1250) — compile-verified
//
#include <hip/hip_runtime.h>

#define IMG 64
#define OUTD 128
#define TILE_H 66
#define TILE_S 72   // floats; image lives at cols 4..67, halo cols 3 and 68, rows 0 and 65

typedef int   v4i __attribute__((ext_vector_type(4)));
typedef float v4f __attribute__((ext_vector_type(4)));
typedef __attribute__((address_space(1))) v4i g_v4i;   // global
typedef __attribute__((address_space(3))) v4i l_v4i;   // LDS

__global__ __launch_bounds__(256) void upfirdn_up2_kernel(
    const float* __restrict__ x, const float* __restrict__ kern,
    float* __restrict__ out)
{
    __shared__ float tile[TILE_H * TILE_S];   // 19008 B per block

    const int tid = threadIdx.x;
    const int img = blockIdx.x;                       // flattened (b*C + c)
    const float* __restrict__ src = x   + (size_t)img * (IMG * IMG);
    float*       __restrict__ dst = out + (size_t)img * (OUTD * OUTD);

    // Phase-decomposed kernel coefficients, kf[a][b] = K[3-a][3-b] (uniform s_loads).
    const float c00_mm = kern[3*4+3], c00_m0 = kern[3*4+1], c00_0m = kern[1*4+3], c00_00 = kern[1*4+1];
    const float c01_m0 = kern[3*4+2], c01_mp = kern[3*4+0], c01_00 = kern[1*4+2], c01_0p = kern[1*4+0];
    const float c10_0m = kern[2*4+3], c10_00 = kern[2*4+1], c10_pm = kern[0*4+3], c10_p0 = kern[0*4+1];
    const float c11_00 = kern[2*4+2], c11_0p = kern[2*4+0], c11_p0 = kern[0*4+2], c11_pp = kern[0*4+0];

    // Phase 1: zero ONLY the 260 halo cells (disjoint from async-written image region,
    // so no barrier needed before issuing the async loads).
    {
        int h = tid;                               // 0..255, then +4 stragglers
        int idx;
        if (h < 66)       idx = 3 + h;                              // top row, cols 3..68
        else if (h < 132) idx = 65 * TILE_S + 3 + (h - 66);         // bottom row
        else if (h < 196) idx = (h - 131) * TILE_S + 3;             // left col, rows 1..64
        else              idx = (h - 195) * TILE_S + 68;            // right col, rows 1..60
        tile[idx] = 0.0f;
        if (tid < 4)
            tile[(61 + tid) * TILE_S + 68] = 0.0f;                  // right col, rows 61..64
    }

    // Phase 2: async-stage the 64x64 image into LDS at tile[r+1][c+4].
    // 1024 float4 segments, 4 per thread; 16B aligned in both spaces.
#if __has_builtin(__builtin_amdgcn_global_load_async_to_lds_b128)
    #pragma unroll
    for (int k = 0; k < 4; ++k) {
        const int seg = tid + k * 256;          // 0..1023
        const int r   = seg >> 4;               // 0..63
        const int c4  = (seg & 15) << 2;        // 0,4,...,60
        const float* g = src + r * IMG + c4;
        float*       l = &tile[(r + 1) * TILE_S + 4 + c4];
        __builtin_amdgcn_global_load_async_to_lds_b128(
            (g_v4i*)g, (l_v4i*)l, /*offset=*/0, /*cpol=*/0);
    }
  #if __has_builtin(__builtin_amdgcn_s_wait_asynccnt)
    __builtin_amdgcn_s_wait_asynccnt(0);
  #else
    asm volatile("s_wait_asynccnt 0" ::: "memory");
  #endif
#else
    #pragma unroll
    for (int k = 0; k < 4; ++k) {
        const int seg = tid + k * 256;
        const int r   = seg >> 4;
        const int c4  = (seg & 15) << 2;
        const v4f v = *(const v4f*)(src + r * IMG + c4);
        *(v4f*)(&tile[(r + 1) * TILE_S + 4 + c4]) = v;
    }
#endif
    __syncthreads();

    // Phase 3: each thread expands 4 groups of 4 consecutive input pixels.
    // Per group: 3 rows x (1 ds_load_b128 + 2 ds_load_b32), 4 NT global_store_b128.
    #pragma unroll
    for (int k = 0; k < 4; ++k) {
        const int grp = tid + k * 256;          // 0..1023
        const int i   = grp >> 4;               // input row 0..63
        const int j4  = (grp & 15) << 2;        // input col 0,4,...,60 (lane-contiguous)
        const float* s = &tile[(i + 1) * TILE_S + 4 + j4];

        const v4f   am = *(const v4f*)(s - TILE_S);
        const float amL = s[-TILE_S - 1], amR = s[-TILE_S + 4];
        const v4f   a0 = *(const v4f*)(s);
        const float a0L = s[-1],          a0R = s[4];
        const v4f   ap = *(const v4f*)(s + TILE_S);
        const float apL = s[TILE_S - 1],  apR = s[TILE_S + 4];

        // Gather 6-wide windows: cols j4-1 .. j4+4 for each of the 3 rows.
        const float wm[6] = { amL, am.x, am.y, am.z, am.w, amR };
        const float w0[6] = { a0L, a0.x, a0.y, a0.z, a0.w, a0R };
        const float wp[6] = { apL, ap.x, ap.y, ap.z, ap.w, apR };

        v4f o0a, o0b, o1a, o1b;   // out rows 2i and 2i+1, cols 2*j4 .. 2*j4+7
        float r0[8], r1[8];
        #pragma unroll
        for (int t = 0; t < 4; ++t) {
            const float mL = wm[t], mC = wm[t + 1], mR = wm[t + 2];
            const float cL = w0[t], cC = w0[t + 1], cR = w0[t + 2];
            const float pL = wp[t], pC = wp[t + 1], pR = wp[t + 2];
            r0[2*t]     = c00_mm * mL + c00_m0 * mC + c00_0m * cL + c00_00 * cC;
            r0[2*t + 1] = c01_m0 * mC + c01_mp * mR + c01_00 * cC + c01_0p * cR;
            r1[2*t]     = c10_0m * cL + c10_00 * cC + c10_pm * pL + c10_p0 * pC;
            r1[2*t + 1] = c11_00 * cC + c11_0p * cR + c11_p0 * pC + c11_pp * pR;
        }
        o0a = (v4f){ r0[0], r0[1], r0[2], r0[3] };
        o0b = (v4f){ r0[4], r0[5], r0[6], r0[7] };
        o1a = (v4f){ r1[0], r1[1], r1[2], r1[3] };
        o1b = (v4f){ r1[4], r1[5], r1[6], r1[7] };

        // Streamed output (537 MB): non-temporal b128 stores, 32B/lane coalesced.
        float* obase0 = dst + (size_t)(2 * i)     * OUTD + 2 * j4;
        float* obase1 = dst + (size_t)(2 * i + 1) * OUTD + 2 * j4;
#if __has_builtin(__builtin_nontemporal_store)
        __builtin_nontemporal_store(o0a, (v4f*)obase0);
        __builtin_nontemporal_store(o0b, (v4f*)(obase0 + 4));
        __builtin_nontemporal_store(o1a, (v4f*)obase1);
        __builtin_nontemporal_store(o1b, (v4f*)(obase1 + 4));
#else
        *(v4f*)obase0       = o0a;
        *(v4f*)(obase0 + 4) = o0b;
        *(v4f*)obase1       = o1a;
        *(v4f*)(obase1 + 4) = o1b;
#endif
    }
}

extern "C" void kernel_launch(void* const* d_in, const int* in_sizes, int n_in,
                              void* d_out, int out_size, void* d_ws, size_t ws_size,
                              hipStream_t stream) {
    const float* x    = (const float*)d_in[0];   // (16, 512, 64, 64) fp32
    const float* kern = (const float*)d_in[1];   // (4, 4) fp32
    float* out = (float*)d_out;                  // (16, 512, 128, 128) fp32

    const int n_imgs = in_sizes[0] / (IMG * IMG);   // 8192
    upfirdn_up2_kernel<<<dim3(n_imgs), dim3(256), 0, stream>>>(x, kern, out);
}